// GCN2_HLBP_23055384445772
// MI455X (gfx1250) — compile-verified
//
#include <hip/hip_runtime.h>
#include <cmath>

#define N_NODES 50000
#define F_IN    500
#define H       64
#define L_LAYERS 8
#define R_RANK  3
#define C_OUT   40
#define E_EDGES 800000
#define ALPHA   0.1f
#define THETA   0.5f
#define EPS_P   4e-3f
#define MIN_NORM 1e-15f

typedef __attribute__((ext_vector_type(2))) float    v2f;
typedef __attribute__((ext_vector_type(8))) float    v8f;
typedef __attribute__((ext_vector_type(4))) unsigned v4u;
typedef __attribute__((ext_vector_type(8))) int      v8i32;
typedef __attribute__((ext_vector_type(4))) int      v4i32;

__device__ __forceinline__ v8f wmma_f32_16x16x4(v2f a, v2f b, v8f c) {
    // 8 args: (neg_a, A, neg_b, B, c_mod, C, reuse_a, reuse_b)
    return __builtin_amdgcn_wmma_f32_16x16x4_f32(false, a, false, b, (short)0, c,
                                                 false, false);
}

// ---------------------------------------------------------------------------
// Kernel A: h0 = relu(x @ W_in + b_in), also initializes h = h0.
// One wave per 16-row tile computing all 4 column tiles (A-fragment reused
// across 4 WMMAs per K-step -> x read exactly once, 100 MB total).
// K-loop kept rolled so the 4 accumulators stay pinned (no v_mov rotation).
// ---------------------------------------------------------------------------
__global__ __launch_bounds__(128) void k_input_gemm(
    const float* __restrict__ x, const float* __restrict__ W_in,
    const float* __restrict__ b_in, float* __restrict__ h0,
    float* __restrict__ h)
{
    int wave = threadIdx.x >> 5;
    int lane = threadIdx.x & 31;
    int laneLo = lane & 15, laneHi = lane >> 4;
    int gw = blockIdx.x * 4 + wave;          // 3125 row tiles
    int tileM = gw * 16;
    if (tileM >= N_NODES) return;
    int row = tileM + laneLo;
    const float* xr = x + (size_t)row * F_IN;

    v8f acc[4] = {};
#pragma unroll 1
    for (int k0 = 0; k0 < F_IN; k0 += 4) {
        int kb = k0 + 2 * laneHi;
        v2f a;
        a.x = xr[kb];
        a.y = xr[kb + 1];
        const float* w0 = W_in + kb * H + laneLo;
        const float* w1 = W_in + (kb + 1) * H + laneLo;
#pragma unroll
        for (int j = 0; j < 4; ++j) {
            v2f b;
            b.x = w0[16 * j];
            b.y = w1[16 * j];
            acc[j] = wmma_f32_16x16x4(a, b, acc[j]);
        }
    }
#pragma unroll
    for (int j = 0; j < 4; ++j) {
        int col = 16 * j + laneLo;
        float bias = b_in[col];
#pragma unroll
        for (int i = 0; i < 8; ++i) {
            int r = tileM + i + 8 * laneHi;
            float v = acc[j][i] + bias;
            v = v > 0.f ? v : 0.f;
            h0[r * H + col] = v;
            h[r * H + col]  = v;
        }
    }
}

// ---------------------------------------------------------------------------
// Kernel B1: SpMM scatter  agg[row] += w * h[col]  via f32 atomics (L2).
// One thread per (edge, 4-feature chunk): 16 threads/edge, float4 gather.
// ---------------------------------------------------------------------------
__global__ __launch_bounds__(256) void k_spmm(
    const int* __restrict__ erow, const int* __restrict__ ecol,
    const float* __restrict__ ew, const float* __restrict__ h,
    float* __restrict__ agg)
{
    int t = blockIdx.x * 256 + threadIdx.x;
    if (t >= E_EDGES * 16) return;
    int e = t >> 4;
    int f = (t & 15) * 4;
    if ((t & 15) == 0) {
        int ep = e + 2048;
        if (ep < E_EDGES) {             // stream edge indices ahead (global_prefetch)
            __builtin_prefetch(&ecol[ep], 0, 0);
            __builtin_prefetch(&erow[ep], 0, 0);
            __builtin_prefetch(&ew[ep], 0, 0);
        }
    }
    int c = ecol[e];
    int r = erow[e];
    float w = ew[e];
    const float4 hv = *(const float4*)(h + (size_t)c * H + f);
    float* dst = agg + (size_t)r * H + f;
    atomicAdd(dst + 0, w * hv.x);
    atomicAdd(dst + 1, w * hv.y);
    atomicAdd(dst + 2, w * hv.z);
    atomicAdd(dst + 3, w * hv.w);
}

// ---------------------------------------------------------------------------
// Kernel B2: sup = (1-a)*agg + a*h0 ; h = relu((1-b)*sup + b*(sup @ Wl)).
// One wave per 16-row tile, all 4 col tiles; sup A-fragments built on the fly
// and reused across 4 WMMAs per K-step. K-loop rolled (pinned accumulators).
// ---------------------------------------------------------------------------
__global__ __launch_bounds__(128) void k_layer_gemm(
    const float* __restrict__ agg, const float* __restrict__ h0,
    const float* __restrict__ Wl, float beta, float* __restrict__ h)
{
    const float ca = ALPHA, c1a = 1.f - ALPHA;
    int wave = threadIdx.x >> 5;
    int lane = threadIdx.x & 31;
    int laneLo = lane & 15, laneHi = lane >> 4;
    int gw = blockIdx.x * 4 + wave;
    int tileM = gw * 16;
    if (tileM >= N_NODES) return;
    int row = tileM + laneLo;

    v8f acc[4] = {};
#pragma unroll 1
    for (int k0 = 0; k0 < H; k0 += 4) {
        int kb = k0 + 2 * laneHi;
        v2f a;
        a.x = c1a * agg[row * H + kb]     + ca * h0[row * H + kb];
        a.y = c1a * agg[row * H + kb + 1] + ca * h0[row * H + kb + 1];
        const float* w0 = Wl + kb * H + laneLo;
        const float* w1 = Wl + (kb + 1) * H + laneLo;
#pragma unroll
        for (int j = 0; j < 4; ++j) {
            v2f b;
            b.x = w0[16 * j];
            b.y = w1[16 * j];
            acc[j] = wmma_f32_16x16x4(a, b, acc[j]);
        }
    }
#pragma unroll
    for (int j = 0; j < 4; ++j) {
        int col = 16 * j + laneLo;
#pragma unroll
        for (int i = 0; i < 8; ++i) {
            int r = tileM + i + 8 * laneHi;
            float sup = c1a * agg[r * H + col] + ca * h0[r * H + col];
            float v = (1.f - beta) * sup + beta * acc[j][i];
            h[r * H + col] = v > 0.f ? v : 0.f;
        }
    }
}

// ---------------------------------------------------------------------------
// Kernel C: fused bilinear pooling + output GEMM + expmap0/proj/log_softmax.
// Block = 128 threads handles 16 nodes. The 16-node h tile is staged to LDS
// by the Tensor Data Mover (TDM); hu/hv live in LDS; the 16x4096 h_bp tile is
// NEVER materialized: A-fragments are synthesized per WMMA step from
// register-cached hv (h_bp[n, m*64+l] = sum_r hu[r,n,m]*hv[r,n,l]), and
// B-fragments stream W_out (L2-resident, 655 KB) with clamped in-bounds
// column indices. 3 waves cover the 40 output classes (3 x 16 cols).
// ---------------------------------------------------------------------------
__global__ __launch_bounds__(128) void k_pool_out(
    const float* __restrict__ h, const float* __restrict__ U_W,
    const float* __restrict__ U_b, const float* __restrict__ V_W,
    const float* __restrict__ V_b, const float* __restrict__ W_out,
    const float* __restrict__ b_out, float* __restrict__ out)
{
    __shared__ float sh_h[16][H];             // 4 KB
    __shared__ float sh_u[R_RANK][16][H];     // 12 KB
    __shared__ float sh_v[R_RANK][16][H];     // 12 KB
    __shared__ float sh_lg[16][48];           // 3 KB (40 cols padded to 48)
    int tid = threadIdx.x;
    int node0 = blockIdx.x * 16;

    // phase 1: TDM-stage the 16x64 f32 h tile into LDS (one DMA, wave 0 only).
    // D# per CDNA5 ISA ch.8: 2-D tensor (dims 0/1), groups 2/3/extra zero.
    if (tid < 32) {
        unsigned lds_off = (unsigned)(unsigned long long)(&sh_h[0][0]);
        unsigned long long ga = (unsigned long long)(h + (size_t)node0 * H);
        v4u g0;
        g0[0] = 1u;                                    // count=1 (valid user D#)
        g0[1] = lds_off;                               // lds_addr (bytes)
        g0[2] = (unsigned)(ga & 0xFFFFFFFFu);          // global_addr[31:0]
        g0[3] = (unsigned)((ga >> 32) & 0x1FFFFFFu)    // global_addr[56:32]
              | (2u << 30);                            // type=2 ("image")
        v8i32 g1;
        g1[0] = (int)(2u << 16);    // wg_mask=0, data_size=2 (4 bytes)
        g1[1] = (int)(64u << 16);   // tensor_dim0[15:0]=64 (row length)
        g1[2] = (int)(16u << 16);   // tensor_dim0[31:16]=0, tensor_dim1[15:0]=16
        g1[3] = (int)(64u << 16);   // tensor_dim1[31:16]=0, tile_dim0=64
        g1[4] = 16;                 // tile_dim1=16, tile_dim2=0
        g1[5] = 64;                 // tensor_dim0_stride=64 (low 32)
        g1[6] = 0;                  // stride0 hi / tensor_dim1_stride lo
        g1[7] = 0;                  // tensor_dim1_stride hi
        v4i32 gz4 = {0, 0, 0, 0};
        v8i32 gz8 = {0, 0, 0, 0, 0, 0, 0, 0};
        // clang-23 / therock form: 6 args (g0, g1, g2, g3, g4, cpol)
        __builtin_amdgcn_tensor_load_to_lds(g0, g1, gz4, gz4, gz8, 0);
        __builtin_amdgcn_s_wait_tensorcnt(0);
    }
    __syncthreads();

    // phase 2: hu[r,n,d] = h[n,:] . U_W[r,:,d] + U_b[r,d]   (same for V)
    for (int o = tid; o < 2 * R_RANK * 16 * H; o += 128) {
        int mat = o / (R_RANK * 16 * H);
        int rem = o % (R_RANK * 16 * H);
        int r = rem / (16 * H);
        int n = (rem / H) & 15;
        int d = rem & 63;
        const float* W = mat ? V_W : U_W;
        const float* B = mat ? V_b : U_b;
        float v = B[r * H + d];
        const float* Wr = W + (size_t)r * H * H + d;
        const float* hn = sh_h[n];
#pragma unroll 8
        for (int k = 0; k < H; ++k) v += hn[k] * Wr[k * H];
        if (mat) sh_v[r][n][d] = v; else sh_u[r][n][d] = v;
    }
    __syncthreads();

    // phase 3: logits tile = h_bp(16x4096) @ W_out(4096x40).
    // hv register-cached (96 VGPRs, m-invariant); 16 WMMAs per m iteration.
    int wave = tid >> 5, lane = tid & 31;
    int laneLo = lane & 15, laneHi = lane >> 4;
    if (wave < 3) {
        int c0 = wave * 16;
        int cc = c0 + laneLo;
        bool cok = cc < C_OUT;
        int ccl = cok ? cc : (C_OUT - 1);    // clamped: loads always in-bounds
        float hva[R_RANK][32];
#pragma unroll
        for (int s = 0; s < 16; ++s) {
            int lb = 4 * s + 2 * laneHi;
#pragma unroll
            for (int r = 0; r < R_RANK; ++r) {
                hva[r][2 * s]     = sh_v[r][laneLo][lb];
                hva[r][2 * s + 1] = sh_v[r][laneLo][lb + 1];
            }
        }
        v8f acc = {};
#pragma unroll 1
        for (int m = 0; m < H; ++m) {
            float au0 = sh_u[0][laneLo][m];
            float au1 = sh_u[1][laneLo][m];
            float au2 = sh_u[2][laneLo][m];
            const float* wo = W_out + (size_t)(m * H + 2 * laneHi) * C_OUT + ccl;
#pragma unroll
            for (int s = 0; s < 16; ++s) {
                v2f a, b;
                a.x = au0 * hva[0][2 * s]     + au1 * hva[1][2 * s]
                    + au2 * hva[2][2 * s];
                a.y = au0 * hva[0][2 * s + 1] + au1 * hva[1][2 * s + 1]
                    + au2 * hva[2][2 * s + 1];
                b.x = wo[(4 * s)     * C_OUT];
                b.y = wo[(4 * s + 1) * C_OUT];
                acc = wmma_f32_16x16x4(a, b, acc);
            }
        }
#pragma unroll
        for (int i = 0; i < 8; ++i) {
            int nloc = i + 8 * laneHi;
            sh_lg[nloc][cc] = acc[i] + b_out[ccl];   // cols >= 40 are never read
        }
    }
    __syncthreads();

    // phase 4: expmap0 -> proj -> log_softmax (c = 1), one thread per node
    if (tid < 16) {
        float y[C_OUT];
        float n2 = 0.f;
        for (int c = 0; c < C_OUT; ++c) { float v = sh_lg[tid][c]; y[c] = v; n2 += v * v; }
        float nrm = sqrtf(n2); nrm = nrm > MIN_NORM ? nrm : MIN_NORM;
        float sc = tanhf(nrm) / nrm;               // expmap0 scale (sqrt_c = 1)
        float yn2 = 0.f;
        for (int c = 0; c < C_OUT; ++c) { y[c] *= sc; yn2 += y[c] * y[c]; }
        float ynorm = sqrtf(yn2); ynorm = ynorm > MIN_NORM ? ynorm : MIN_NORM;
        const float maxn = 1.f - EPS_P;
        if (ynorm > maxn) {
            float s = maxn / ynorm;
            for (int c = 0; c < C_OUT; ++c) y[c] *= s;
        }
        float mx = -1e30f;
        for (int c = 0; c < C_OUT; ++c) mx = y[c] > mx ? y[c] : mx;
        float se = 0.f;
        for (int c = 0; c < C_OUT; ++c) se += expf(y[c] - mx);
        float lse = logf(se);
        float* op = out + (size_t)(node0 + tid) * C_OUT;
        for (int c = 0; c < C_OUT; ++c) op[c] = y[c] - mx - lse;
    }
}

// ---------------------------------------------------------------------------
extern "C" void kernel_launch(void* const* d_in, const int* in_sizes, int n_in,
                              void* d_out, int out_size, void* d_ws, size_t ws_size,
                              hipStream_t stream) {
    const float* x      = (const float*)d_in[0];
    const int*   erow   = (const int*)d_in[1];
    const int*   ecol   = (const int*)d_in[2];
    const float* ew     = (const float*)d_in[3];
    const float* W_in   = (const float*)d_in[4];
    const float* b_in   = (const float*)d_in[5];
    const float* conv_W = (const float*)d_in[6];
    const float* U_W    = (const float*)d_in[7];
    const float* U_b    = (const float*)d_in[8];
    const float* V_W    = (const float*)d_in[9];
    const float* V_b    = (const float*)d_in[10];
    const float* W_out  = (const float*)d_in[11];
    const float* b_out  = (const float*)d_in[12];
    float* out = (float*)d_out;

    const size_t NH = (size_t)N_NODES * H;   // 3.2M floats = 12.8 MB
    float* ws  = (float*)d_ws;
    float* h0  = ws;                         // [0,  NH)
    float* h   = ws + NH;                    // [NH, 2NH)
    float* agg = ws + 2 * NH;                // [2NH, 3NH)

    const int nTiles = N_NODES / 16;                       // 3125
    const int gemmBlocks = (nTiles + 3) / 4;               // 782 (4 waves/block)

    k_input_gemm<<<gemmBlocks, 128, 0, stream>>>(x, W_in, b_in, h0, h);

    for (int l = 0; l < L_LAYERS; ++l) {
        (void)hipMemsetAsync(agg, 0, NH * sizeof(float), stream);
        k_spmm<<<(E_EDGES * 16 + 255) / 256, 256, 0, stream>>>(erow, ecol, ew, h, agg);
        float beta = logf(THETA / (float)(l + 1) + 1.0f);
        k_layer_gemm<<<gemmBlocks, 128, 0, stream>>>(
            agg, h0, conv_W + (size_t)l * H * H, beta, h);
    }

    k_pool_out<<<nTiles, 128, 0, stream>>>(h, U_W, U_b, V_W, V_b,
                                           W_out, b_out, out);
}